// EmbeddingDropout_15152644620959
// MI455X (gfx1250) — compile-verified
//
#include <hip/hip_runtime.h>
#include <stdint.h>

// ---------------------------------------------------------------------------
// EmbeddingDropout: out[tok, :] = threefry_mask(row) * weight[row, :] / 0.9
//   words : [16384] int   (B=8, S=2048)
//   weight: [50257, 1024] float
//   out   : [16384, 1024] float
//
// Memory-bound gather (no matmul -> no WMMA). CDNA5 paths used:
//   * global_load_async_to_lds_b128 + s_wait_asynccnt double-buffered row
//     pipeline (gather latency hidden by the async copy engine)
//   * ds_load_b128 from LDS, non-temporal global_store_b128 for the
//     write-once output (keeps the ~196 MiB table resident in the 192 MB L2)
// Dropout mask = exact JAX threefry-2x32 (20 rounds), key (0,42), recomputed
// on the fly from the flat table index -> zero mask-memory traffic.
// ---------------------------------------------------------------------------

typedef float v4f __attribute__((ext_vector_type(4)));
typedef int   v4i __attribute__((ext_vector_type(4)));
// Pointer types matching the async-to-LDS builtin's prototype:
//   param0: int4 in addrspace(1) (global), param1: int4 in addrspace(3) (LDS)
typedef __attribute__((address_space(1))) v4i* gptr_v4i;
typedef __attribute__((address_space(3))) v4i* lptr_v4i;

namespace {
constexpr int      kD             = 1024;          // embedding dim
constexpr int      kWavesPerBlock = 4;             // 128 threads
constexpr int      kThreads       = kWavesPerBlock * 32;
constexpr int      kTokPerWave    = 4;
constexpr int      kTokPerBlock   = kWavesPerBlock * kTokPerWave; // 16
constexpr uint32_t kHalf          = 25731584u;     // (50257*1024)/2
}

#if __has_builtin(__builtin_amdgcn_global_load_async_to_lds_b128)
#define HAVE_ASYNC_LDS 1
#else
#define HAVE_ASYNC_LDS 0
#warning "gfx1250 async-to-LDS builtin not found; compiling direct-gather fallback"
#endif

#if __has_builtin(__builtin_amdgcn_s_wait_asynccnt)
#define WAIT_ASYNC(n) __builtin_amdgcn_s_wait_asynccnt(n)
#else
#define WAIT_ASYNC(n) asm volatile("s_wait_asynccnt %0" ::"i"(n) : "memory")
#endif

__device__ __forceinline__ uint32_t rotl32(uint32_t x, uint32_t r) {
  return (x << r) | (x >> (32u - r));
}

// Exact JAX threefry-2x32-20 for table flat-index j, key = (0, 42).
// JAX splits counters [0,n) into halves: draw(j) = x0 of pair (j, j+n/2) for
// j < n/2, else x1 of pair (j-n/2, j). uniform = bits>>9 | 0x3f800000 - 1.0,
// keep iff uniform < 0.9.  Returns the dropout multiplier (1/0.9 or 0).
__device__ __forceinline__ float dropout_scale(uint32_t j) {
  const uint32_t ks1 = 42u;
  const uint32_t ks2 = 42u ^ 0x1BD11BDAu;
  const bool     lo  = j < kHalf;
  const uint32_t c0  = lo ? j : (j - kHalf);
  uint32_t x0 = c0;                    // + ks0 (== 0)
  uint32_t x1 = (c0 + kHalf) + ks1;
#define TF_QR(r)        \
  x0 += x1;             \
  x1 = rotl32(x1, r);   \
  x1 ^= x0;
  TF_QR(13) TF_QR(15) TF_QR(26) TF_QR(6)   x0 += ks1;  x1 += ks2 + 1u;
  TF_QR(17) TF_QR(29) TF_QR(16) TF_QR(24)  x0 += ks2;  x1 += 2u;          // ks0==0
  TF_QR(13) TF_QR(15) TF_QR(26) TF_QR(6)   /*x0+=0*/   x1 += ks1 + 3u;
  TF_QR(17) TF_QR(29) TF_QR(16) TF_QR(24)  x0 += ks1;  x1 += ks2 + 4u;
  TF_QR(13) TF_QR(15) TF_QR(26) TF_QR(6)   x0 += ks2;  x1 += 5u;          // ks0==0
#undef TF_QR
  const uint32_t bits = lo ? x0 : x1;
  const float u = __uint_as_float((bits >> 9) | 0x3f800000u) - 1.0f;
  // reference does weight/0.9; multiply by the rounded reciprocal (<=1 ulp off)
  return (u < 0.9f) ? (1.0f / 0.9f) : 0.0f;
}

__device__ __forceinline__ void mask_store(v4f v, uint32_t j, v4f* dst) {
  v4f r;
  r.x = v.x * dropout_scale(j + 0u);
  r.y = v.y * dropout_scale(j + 1u);
  r.z = v.z * dropout_scale(j + 2u);
  r.w = v.w * dropout_scale(j + 3u);
  __builtin_nontemporal_store(r, dst);   // write-once output: TH=NT
}

__global__ __launch_bounds__(kThreads) void embdrop_gather_kernel(
    const int* __restrict__ words, const float* __restrict__ weight,
    float* __restrict__ out, int n_tokens) {
  const int lane     = (int)(threadIdx.x & 31u);
  const int wave     = (int)(threadIdx.x >> 5u);
  const int tok_base = ((int)blockIdx.x * kWavesPerBlock + wave) * kTokPerWave;

#if HAVE_ASYNC_LDS
  // 4 KiB row buffer x 2 (double buffer) x 4 waves = 32 KiB
  __shared__ __align__(16) float lds[kWavesPerBlock * 2 * kD];
  float* const buf0 = &lds[(wave * 2 + 0) * kD];
  float* const buf1 = &lds[(wave * 2 + 1) * kD];

  // --- prologue: issue async gather of token 0's row ------------------------
  int w_cur = 0;
  if (tok_base < n_tokens) {
    w_cur = words[tok_base];
    const float* src = weight + (size_t)(uint32_t)w_cur * kD;
#pragma unroll
    for (int k = 0; k < 8; ++k) {
      __builtin_amdgcn_global_load_async_to_lds_b128(
          (gptr_v4i)(src + (lane + k * 32) * 4),
          (lptr_v4i)(buf0 + (lane + k * 32) * 4), 0, 0);
    }
  }

  // --- steady state: issue row t+1, wait for row t, mask+store row t --------
  for (int t = 0; t < kTokPerWave; ++t) {
    const int tok = tok_base + t;
    if (tok >= n_tokens) break;

    int w_next = 0;
    const bool have_next = (t + 1 < kTokPerWave) && (tok + 1 < n_tokens);
    if (have_next) {
      w_next = words[tok + 1];
      const float* src = weight + (size_t)(uint32_t)w_next * kD;
      float* nb = ((t + 1) & 1) ? buf1 : buf0;
#pragma unroll
      for (int k = 0; k < 8; ++k) {
        __builtin_amdgcn_global_load_async_to_lds_b128(
            (gptr_v4i)(src + (lane + k * 32) * 4),
            (lptr_v4i)(nb + (lane + k * 32) * 4), 0, 0);
      }
      WAIT_ASYNC(8);   // 8 outstanding = next row; current row's 8 are done
    } else {
      WAIT_ASYNC(0);
    }
    asm volatile("" ::: "memory");  // don't hoist LDS reads above the wait

    const float*   rb      = (t & 1) ? buf1 : buf0;
    const uint32_t rowbase = (uint32_t)w_cur * (uint32_t)kD;
    v4f*           outp    = (v4f*)(out + (size_t)tok * kD);
#pragma unroll
    for (int k = 0; k < 8; ++k) {
      const int f4 = lane + k * 32;
      v4f v = *(const v4f*)(rb + f4 * 4);           // ds_load_b128
      mask_store(v, rowbase + (uint32_t)(f4 * 4), outp + f4);
    }
    w_cur = w_next;
  }
#else
  // ---- fallback: direct b128 gather -> mask -> NT store --------------------
  for (int t = 0; t < kTokPerWave; ++t) {
    const int tok = tok_base + t;
    if (tok >= n_tokens) break;
    const int      w       = words[tok];
    const v4f*     srow    = (const v4f*)(weight + (size_t)(uint32_t)w * kD);
    const uint32_t rowbase = (uint32_t)w * (uint32_t)kD;
    v4f*           outp    = (v4f*)(out + (size_t)tok * kD);
#pragma unroll
    for (int k = 0; k < 8; ++k) {
      const int f4 = lane + k * 32;
      v4f v = srow[f4];
      mask_store(v, rowbase + (uint32_t)(f4 * 4), outp + f4);
    }
  }
#endif
}

extern "C" void kernel_launch(void* const* d_in, const int* in_sizes, int n_in,
                              void* d_out, int out_size, void* d_ws, size_t ws_size,
                              hipStream_t stream) {
  const int*   words  = (const int*)d_in[0];    // [B*S] token ids
  const float* weight = (const float*)d_in[1];  // [V, 1024]
  float*       out    = (float*)d_out;          // [B*S, 1024]
  const int n_tokens  = in_sizes[0];            // 16384
  const int blocks    = (n_tokens + kTokPerBlock - 1) / kTokPerBlock;  // 1024
  hipLaunchKernelGGL(embdrop_gather_kernel, dim3(blocks), dim3(kThreads), 0,
                     stream, words, weight, out, n_tokens);
}